// TOPKPOOL_AE_32392643346841
// MI455X (gfx1250) — compile-verified
//
#include <hip/hip_runtime.h>
#include <hip/hip_bf16.h>
#include <stdint.h>

typedef float v2f __attribute__((ext_vector_type(2)));
typedef float v8f __attribute__((ext_vector_type(8)));

#define FDIM 128
#define AS_STRIDE 132   // 16x128 A tile padded to avoid LDS bank conflicts
#define NBINS 2048

// ---------------------------------------------------------------------------
// GEMM: H[N x 128] = X[N x 128] @ W[128 x 128], fp32 via V_WMMA_F32_16X16X4_F32
// Block: 256 threads = 8 waves; block handles 16 rows, wave w handles cols [16w,16w+16)
// A tile staged to LDS via GLOBAL_LOAD_ASYNC_TO_LDS_B128 (ASYNCcnt path).
// ---------------------------------------------------------------------------
__global__ __launch_bounds__(256) void gemm128_wmma(const float* __restrict__ X,
                                                    const float* __restrict__ Wm,
                                                    float* __restrict__ H, int N) {
    __shared__ float As[16 * AS_STRIDE];
    const int tid   = threadIdx.x;
    const int rbase = blockIdx.x * 16;

    // prefetch W (hot in L2, pull toward WGP): 256 threads x 256B covers 64KB
    __builtin_prefetch(&Wm[(size_t)tid * 64], 0, 0);

    // stage A tile (16x128) into LDS via async DMA loads, clamp OOB rows
    for (int idx = tid * 4; idx < 16 * FDIM; idx += 256 * 4) {
        int r = idx >> 7, cc = idx & 127;
        int row = rbase + r;
        const float* gp = &X[(size_t)(row < N ? row : 0) * FDIM + cc];
        unsigned int lds_off = (unsigned int)(uintptr_t)(&As[r * AS_STRIDE + cc]);
        unsigned long long ga = (unsigned long long)(uintptr_t)gp;
        asm volatile("global_load_async_to_lds_b128 %0, %1, off"
                     :: "v"(lds_off), "v"(ga)
                     : "memory");
    }
    asm volatile("s_wait_asynccnt 0" ::: "memory");
    __syncthreads();

    const int wave  = tid >> 5;         // column tile 0..7
    const int lane  = tid & 31;
    const int m     = lane & 15;        // row within tile (A) / col within tile (B,C)
    const int khalf = (lane >> 4) * 2;  // lanes 0-15 -> K{0,1}; lanes 16-31 -> K{2,3}
    const int ncol  = wave * 16 + m;

    v8f c = {0.f, 0.f, 0.f, 0.f, 0.f, 0.f, 0.f, 0.f};
    const float* wp = Wm + ncol;
#pragma unroll 4
    for (int k = 0; k < FDIM; k += 4) {
        v2f a;
        a.x = As[m * AS_STRIDE + k + khalf];
        a.y = As[m * AS_STRIDE + k + khalf + 1];
        v2f b;
        b.x = wp[(size_t)(k + khalf) * FDIM];
        b.y = wp[(size_t)(k + khalf + 1) * FDIM];
        c = __builtin_amdgcn_wmma_f32_16x16x4_f32(false, a, false, b,
                                                  (short)0, c, false, false);
    }

    // C/D layout: VGPR i -> (lanes 0-15: row i, col lane) (lanes 16-31: row 8+i, col lane-16)
    const int rowoff = (lane >> 4) * 8;
    if (rbase + 16 <= N) {
        // fast path: straight-line stores with immediate offsets
        float* hp = &H[(size_t)(rbase + rowoff) * FDIM + ncol];
#pragma unroll
        for (int i = 0; i < 8; ++i) hp[(size_t)i * FDIM] = c[i];
    } else {
#pragma unroll
        for (int i = 0; i < 8; ++i) {
            int row = rbase + rowoff + i;
            if (row < N) H[(size_t)row * FDIM + ncol] = c[i];
        }
    }
}

// ---------------------------------------------------------------------------
// Degree accumulation (edge weight 1.0 at target), original edges
// ---------------------------------------------------------------------------
__global__ __launch_bounds__(256) void deg_ones(const int* __restrict__ cols,
                                                float* __restrict__ deg, int E) {
    int e = blockIdx.x * blockDim.x + threadIdx.x;
    if (e >= E) return;
    atomicAdd(&deg[cols[e]], 1.0f);
}

// Degree for kept (remapped) edges after pooling
__global__ __launch_bounds__(256) void deg_kept(const int* __restrict__ rows,
                                                const int* __restrict__ cols,
                                                const int* __restrict__ newidx,
                                                float* __restrict__ deg, int E) {
    int e = blockIdx.x * blockDim.x + threadIdx.x;
    if (e >= E) return;
    int nr = newidx[rows[e]], nc = newidx[cols[e]];
    if (nr >= 0 && nc >= 0) atomicAdd(&deg[nc], 1.0f);
}

__global__ __launch_bounds__(256) void to_dinv(float* __restrict__ d, int N) {
    int i = blockIdx.x * blockDim.x + threadIdx.x;
    if (i < N) d[i] = rsqrtf(d[i] + 1.0f);   // deg includes +1 self loop
}

// ---------------------------------------------------------------------------
// Edge scatter: AG[col] += dinv[row]*dinv[col] * H[row]. One wave per edge,
// each lane handles 4 features (float4 gather + 4 f32 atomics to L2).
// Edge index is wave-uniform: force scalar (SMEM) loads via readfirstlane.
// ---------------------------------------------------------------------------
__global__ __launch_bounds__(256) void scatter_ones(const int* __restrict__ rows,
                                                    const int* __restrict__ cols,
                                                    const float* __restrict__ dinv,
                                                    const float* __restrict__ H,
                                                    float* __restrict__ AG, int E) {
    long long t = (long long)blockIdx.x * blockDim.x + threadIdx.x;
    int e = (int)(t >> 5);
    if (e >= E) return;                              // wave-uniform guard
    e = __builtin_amdgcn_readfirstlane(e);
    int lane = (int)t & 31;
    int r = rows[e], c = cols[e];
    float nrm = dinv[r] * dinv[c];
    float4 h = *(const float4*)&H[(size_t)r * FDIM + lane * 4];
    float* o = &AG[(size_t)c * FDIM + lane * 4];
    atomicAdd(o + 0, nrm * h.x);
    atomicAdd(o + 1, nrm * h.y);
    atomicAdd(o + 2, nrm * h.z);
    atomicAdd(o + 3, nrm * h.w);
}

__global__ __launch_bounds__(256) void scatter_kept(const int* __restrict__ rows,
                                                    const int* __restrict__ cols,
                                                    const int* __restrict__ newidx,
                                                    const float* __restrict__ dinv,
                                                    const float* __restrict__ H,
                                                    float* __restrict__ AG, int E) {
    long long t = (long long)blockIdx.x * blockDim.x + threadIdx.x;
    int e = (int)(t >> 5);
    if (e >= E) return;                              // wave-uniform guard
    e = __builtin_amdgcn_readfirstlane(e);
    int lane = (int)t & 31;
    int nr = newidx[rows[e]], nc = newidx[cols[e]];
    if (nr < 0 || nc < 0) return;   // masked edge: weight 0, true no-op (uniform)
    float nrm = dinv[nr] * dinv[nc];
    float4 h = *(const float4*)&H[(size_t)nr * FDIM + lane * 4];
    float* o = &AG[(size_t)nc * FDIM + lane * 4];
    atomicAdd(o + 0, nrm * h.x);
    atomicAdd(o + 1, nrm * h.y);
    atomicAdd(o + 2, nrm * h.z);
    atomicAdd(o + 3, nrm * h.w);
}

// ---------------------------------------------------------------------------
// Finalize (in place on OUT which holds the aggregate):
//   OUT = act( AG + dinv^2 * H + bias ),  act = tanh or identity
// ---------------------------------------------------------------------------
__global__ __launch_bounds__(256) void finalize_kernel(float* __restrict__ OUT,
                                                       const float* __restrict__ H,
                                                       const float* __restrict__ dinv,
                                                       const float* __restrict__ bias,
                                                       int N, int act) {
    int t = blockIdx.x * blockDim.x + threadIdx.x;   // one float4 per thread
    if (t >= N * 32) return;
    int i = t >> 5;
    int f = (t & 31) * 4;
    float di = dinv[i];
    float s = di * di;
    float4 h = *(const float4*)&H[(size_t)i * FDIM + f];
    float4 a = *(const float4*)&OUT[(size_t)i * FDIM + f];
    float4 bb = *(const float4*)&bias[f];
    float4 o;
    o.x = a.x + s * h.x + bb.x;
    o.y = a.y + s * h.y + bb.y;
    o.z = a.z + s * h.z + bb.z;
    o.w = a.w + s * h.w + bb.w;
    if (act) { o.x = tanhf(o.x); o.y = tanhf(o.y); o.z = tanhf(o.z); o.w = tanhf(o.w); }
    *(float4*)&OUT[(size_t)i * FDIM + f] = o;
}

// ---------------------------------------------------------------------------
// Pooling: 1/||w||
// ---------------------------------------------------------------------------
__global__ void norm_kernel(const float* __restrict__ pw, float* __restrict__ rn) {
    __shared__ float sm[FDIM];
    int t = threadIdx.x;
    float v = pw[t];
    sm[t] = v * v;
    __syncthreads();
    for (int s = 64; s > 0; s >>= 1) {
        if (t < s) sm[t] += sm[t + s];
        __syncthreads();
    }
    if (t == 0) rn[0] = rsqrtf(sm[0]);
}

// score[i] = tanh((x[i].w)/||w||), plus bin histogram. One wave per node.
__global__ __launch_bounds__(256) void score_kernel(const float* __restrict__ X,
                                                    const float* __restrict__ pw,
                                                    const float* __restrict__ rn,
                                                    float* __restrict__ score,
                                                    int* __restrict__ binIdx,
                                                    int* __restrict__ hist, int N) {
    int wid  = (blockIdx.x * blockDim.x + threadIdx.x) >> 5;
    int lane = threadIdx.x & 31;
    if (wid >= N) return;
    float4 xv = *(const float4*)&X[(size_t)wid * FDIM + lane * 4];
    float4 wv = *(const float4*)&pw[lane * 4];
    float d = xv.x * wv.x + xv.y * wv.y + xv.z * wv.z + xv.w * wv.w;
    for (int o = 16; o > 0; o >>= 1) d += __shfl_down(d, o, 32);
    if (lane == 0) {
        float s = tanhf(d * rn[0]);
        score[wid] = s;
        int b = (int)((s + 1.0f) * 1024.0f);
        b = b < 0 ? 0 : (b > NBINS - 1 ? NBINS - 1 : b);
        binIdx[wid] = b;
        atomicAdd(&hist[b], 1);
    }
}

__global__ void prefix_kernel(const int* __restrict__ hist, int* __restrict__ off) {
    if (blockIdx.x == 0 && threadIdx.x == 0) {
        int run = 0;
        for (int b = 0; b < NBINS; ++b) { off[b] = run; run += hist[b]; }
    }
}

__global__ __launch_bounds__(256) void fill_kernel(const int* __restrict__ binIdx,
                                                   const int* __restrict__ off,
                                                   int* __restrict__ fillc,
                                                   int* __restrict__ binList, int N) {
    int i = blockIdx.x * blockDim.x + threadIdx.x;
    if (i >= N) return;
    int b = binIdx[i];
    int pos = off[b] + atomicAdd(&fillc[b], 1);
    binList[pos] = i;
}

// Exact descending rank (tie-break: lower index first, like jax top_k).
// rank = (#elements in strictly-higher bins) + exact count within own bin.
__global__ __launch_bounds__(256) void rank_kernel(const float* __restrict__ score,
                                                   const int* __restrict__ binIdx,
                                                   const int* __restrict__ off,
                                                   const int* __restrict__ hist,
                                                   const int* __restrict__ binList,
                                                   int* __restrict__ newidx,
                                                   int N, int K) {
    int i = blockIdx.x * blockDim.x + threadIdx.x;
    if (i >= N) return;
    int b = binIdx[i];
    float si = score[i];
    int start = off[b], cnt = hist[b];
    int rank = N - (start + cnt);           // everything in higher bins outranks i
    for (int j = 0; j < cnt; ++j) {
        int idx = binList[start + j];
        float sj = score[idx];
        if (sj > si || (sj == si && idx < i)) rank++;
    }
    newidx[i] = (rank < K) ? rank : -1;
}

// x_out[i] = selected(i) ? x[i]*score[i] : 0   (scatter of x_pool back at perm)
__global__ __launch_bounds__(256) void xout_kernel(const float* __restrict__ X,
                                                   const float* __restrict__ score,
                                                   const int* __restrict__ newidx,
                                                   float* __restrict__ XO, int N) {
    int t = blockIdx.x * blockDim.x + threadIdx.x;   // one float4 per thread
    if (t >= N * 32) return;
    int i = t >> 5;
    float s = (newidx[i] >= 0) ? score[i] : 0.0f;
    float4 v = *(const float4*)&X[(size_t)t * 4];
    v.x *= s; v.y *= s; v.z *= s; v.w *= s;
    *(float4*)&XO[(size_t)t * 4] = v;
}

// ---------------------------------------------------------------------------
extern "C" void kernel_launch(void* const* d_in, const int* in_sizes, int n_in,
                              void* d_out, int out_size, void* d_ws, size_t ws_size,
                              hipStream_t stream) {
    const int N = in_sizes[0] / FDIM;
    const int E = in_sizes[1] / 2;
    const int K = (N + 1) / 2;               // ceil(0.5 * N)
    const size_t NF = (size_t)N * FDIM;

    const float* nodes = (const float*)d_in[0];
    const int*   rows  = (const int*)d_in[1];
    const int*   cols  = rows + E;
    const float* W1 = (const float*)d_in[3];  const float* b1 = (const float*)d_in[4];
    const float* W2 = (const float*)d_in[5];  const float* b2 = (const float*)d_in[6];
    const float* pw = (const float*)d_in[7];
    const float* W3 = (const float*)d_in[8];  const float* b3 = (const float*)d_in[9];
    const float* W4 = (const float*)d_in[10]; const float* b4 = (const float*)d_in[11];
    const float* W5 = (const float*)d_in[12]; const float* b5 = (const float*)d_in[13];
    float* OUT = (float*)d_out;

    // workspace carve (256B aligned)
    uint8_t* base = (uint8_t*)d_ws;
    size_t woff = 0;
    auto carve = [&](size_t bytes) -> void* {
        woff = (woff + 255) & ~(size_t)255;
        void* p = base + woff;
        woff += bytes;
        return p;
    };
    float* A     = (float*)carve(NF * 4);
    float* Bb    = (float*)carve(NF * 4);
    float* H     = (float*)carve(NF * 4);
    float* dinv1 = (float*)carve((size_t)N * 4);
    float* dinv2 = (float*)carve((size_t)N * 4);
    float* score = (float*)carve((size_t)N * 4);
    int*  binIdx = (int*)carve((size_t)N * 4);
    int*  newidx = (int*)carve((size_t)N * 4);
    int* binList = (int*)carve((size_t)N * 4);
    int*    hist = (int*)carve(NBINS * 4);
    int*    offb = (int*)carve(NBINS * 4);
    int*   fillc = (int*)carve(NBINS * 4);
    float*    rn = (float*)carve(256);

    const int gemmBlocks = (N + 15) / 16;
    const int edgeBlocks = (E + 255) / 256;
    const int nodeBlocks = (N + 255) / 256;
    const long long scatT = (long long)E * 32;
    const int scatBlocks = (int)((scatT + 255) / 256);
    const int elemBlocks = (int)(((long long)N * 32 + 255) / 256);

    // ---- dinv for original graph (edge weight 1) ----
    hipMemsetAsync(dinv1, 0, (size_t)N * 4, stream);
    deg_ones<<<edgeBlocks, 256, 0, stream>>>(cols, dinv1, E);
    to_dinv<<<nodeBlocks, 256, 0, stream>>>(dinv1, N);

    // ---- layer 1: nodes -> A (tanh) ----
    gemm128_wmma<<<gemmBlocks, 256, 0, stream>>>(nodes, W1, H, N);
    hipMemsetAsync(A, 0, NF * 4, stream);
    scatter_ones<<<scatBlocks, 256, 0, stream>>>(rows, cols, dinv1, H, A, E);
    finalize_kernel<<<elemBlocks, 256, 0, stream>>>(A, H, dinv1, b1, N, 1);

    // ---- layer 2: A -> Bb (tanh) ----
    gemm128_wmma<<<gemmBlocks, 256, 0, stream>>>(A, W2, H, N);
    hipMemsetAsync(Bb, 0, NF * 4, stream);
    scatter_ones<<<scatBlocks, 256, 0, stream>>>(rows, cols, dinv1, H, Bb, E);
    finalize_kernel<<<elemBlocks, 256, 0, stream>>>(Bb, H, dinv1, b2, N, 1);

    // ---- TopK pooling on Bb ----
    norm_kernel<<<1, 128, 0, stream>>>(pw, rn);
    hipMemsetAsync(hist, 0, NBINS * 4, stream);
    hipMemsetAsync(fillc, 0, NBINS * 4, stream);
    score_kernel<<<(N * 32 + 255) / 256, 256, 0, stream>>>(Bb, pw, rn, score, binIdx, hist, N);
    prefix_kernel<<<1, 1, 0, stream>>>(hist, offb);
    fill_kernel<<<nodeBlocks, 256, 0, stream>>>(binIdx, offb, fillc, binList, N);
    rank_kernel<<<nodeBlocks, 256, 0, stream>>>(score, binIdx, offb, hist, binList, newidx, N, K);
    xout_kernel<<<elemBlocks, 256, 0, stream>>>(Bb, score, newidx, A, N);   // A = x_out

    // ---- dinv for pooled (remapped) graph ----
    hipMemsetAsync(dinv2, 0, (size_t)N * 4, stream);
    deg_kept<<<edgeBlocks, 256, 0, stream>>>(rows, cols, newidx, dinv2, E);
    to_dinv<<<nodeBlocks, 256, 0, stream>>>(dinv2, N);

    // ---- layer 3: A -> Bb (tanh) ----
    gemm128_wmma<<<gemmBlocks, 256, 0, stream>>>(A, W3, H, N);
    hipMemsetAsync(Bb, 0, NF * 4, stream);
    scatter_kept<<<scatBlocks, 256, 0, stream>>>(rows, cols, newidx, dinv2, H, Bb, E);
    finalize_kernel<<<elemBlocks, 256, 0, stream>>>(Bb, H, dinv2, b3, N, 1);

    // ---- layer 4: Bb -> A (tanh) ----
    gemm128_wmma<<<gemmBlocks, 256, 0, stream>>>(Bb, W4, H, N);
    hipMemsetAsync(A, 0, NF * 4, stream);
    scatter_kept<<<scatBlocks, 256, 0, stream>>>(rows, cols, newidx, dinv2, H, A, E);
    finalize_kernel<<<elemBlocks, 256, 0, stream>>>(A, H, dinv2, b4, N, 1);

    // ---- layer 5: A -> OUT (no tanh) ----
    gemm128_wmma<<<gemmBlocks, 256, 0, stream>>>(A, W5, H, N);
    hipMemsetAsync(OUT, 0, NF * 4, stream);
    scatter_kept<<<scatBlocks, 256, 0, stream>>>(rows, cols, newidx, dinv2, H, OUT, E);
    finalize_kernel<<<elemBlocks, 256, 0, stream>>>(OUT, H, dinv2, b5, N, 0);
}